// RexSAGEConv_49357764165687
// MI455X (gfx1250) — compile-verified
//
#include <hip/hip_runtime.h>
#include <hip/hip_bf16.h>

#define N_NODES 50000
#define N_EDGES 800000
#define DIM 64
#define OUT_DIM 40
#define OUT_PAD 48

typedef float v2f __attribute__((ext_vector_type(2)));
typedef float v8f __attribute__((ext_vector_type(8)));

// ---------------- degree count (int atomics: deterministic) ----------------
__global__ __launch_bounds__(256) void deg_kernel(const int* __restrict__ src,
                                                  int* __restrict__ deg, int E) {
    int t = blockIdx.x * blockDim.x + threadIdx.x;
    if (t < E) atomicAdd(&deg[src[t]], 1);
}

// ---------------- fused GEMM via fp32 WMMA -------------------------------
// out[r][n] = bias[n] + sum_k f(A[r][k]) * W[k][n],  W row-major (K x NC)
// f = identity when deg==nullptr, else relu(a / deg[r]) (deg==0 -> 0).
// One 16-row M-tile per block; wave w owns N-columns [16w, 16w+16).
// K=64 consumed as 16 x V_WMMA_F32_16X16X4_F32 accumulations.
template <int NC>
__global__ __launch_bounds__(32 * (NC / 16)) void gemm_wmma(
    const float* __restrict__ A, const float* __restrict__ W,
    const float* __restrict__ bias, const int* __restrict__ deg,
    float* __restrict__ out) {
    const int wid  = threadIdx.x >> 5;
    const int lane = threadIdx.x & 31;
    const int m    = lane & 15;   // M index (A rows) / N index (B cols)
    const int half = lane >> 4;   // selects K-pair {0,1} vs {2,3} within k-step
    const int base = blockIdx.x * 16;
    const int row  = base + m;
    const int n    = wid * 16 + m;

    // Fold mean-normalization + ReLU of the previous aggregation into A load.
    const bool pre = (deg != nullptr);   // uniform: EXEC stays all-1s for WMMA
    float scale = 1.0f;
    if (pre) {
        int d = deg[row];
        scale = (d > 0) ? (1.0f / (float)d) : 0.0f;
    }

    // C init: bias depends only on column n; identical across the 8 acc VGPRs.
    const float bn = bias[n];
    v8f acc;
#pragma unroll
    for (int r = 0; r < 8; ++r) acc[r] = bn;

    const float* Arow = A + (long)row * DIM;

#pragma unroll
    for (int k = 0; k < DIM; k += 4) {
        const int k0 = k + 2 * half;
        // A fragment (16x4 f32): lane m holds (K=k0, K=k0+1) -> contiguous float2
        float2 av = *(const float2*)(Arow + k0);
        float a0 = av.x, a1 = av.y;
        if (pre) {
            a0 = fmaxf(a0 * scale, 0.0f);
            a1 = fmaxf(a1 * scale, 0.0f);
        }
        // B fragment (4x16 f32): column n, rows k0 / k0+1 (stride-NC loads; W is
        // tiny, L0/L2 resident)
        float b0 = W[k0 * NC + n];
        float b1 = W[(k0 + 1) * NC + n];
        v2f a2 = {a0, a1};
        v2f b2 = {b0, b1};
        acc = __builtin_amdgcn_wmma_f32_16x16x4_f32(
            /*neg_a=*/false, a2, /*neg_b=*/false, b2,
            /*c_mod=*/(short)0, acc, /*reuse_a=*/false, /*reuse_b=*/false);
    }

    // D layout: VGPR r -> rows (base+r) [lanes 0-15] and (base+8+r) [lanes 16-31]
#pragma unroll
    for (int r = 0; r < 8; ++r)
        out[(long)(base + half * 8 + r) * NC + n] = acc[r];
}

// ---------------- edge-parallel scatter-add: agg[src] += h[dst] -----------
__global__ __launch_bounds__(256) void scatter_kernel(
    const int* __restrict__ src, const int* __restrict__ dst,
    const float* __restrict__ h, float* __restrict__ agg, int E) {
    long t = (long)blockIdx.x * blockDim.x + threadIdx.x;
    int e = (int)(t >> 4);
    int p = (int)(t & 15);
    if (e >= E) return;
    int s = src[e];
    int d = dst[e];
    const float4 v = *(const float4*)(h + (long)d * DIM + p * 4);
    float* o = agg + (long)s * DIM + p * 4;
    atomicAdd(o + 0, v.x);
    atomicAdd(o + 1, v.y);
    atomicAdd(o + 2, v.z);
    atomicAdd(o + 3, v.w);
}

// ---------------- pad Wp2 (64x40) -> (64x48), bp2 -> 48 -------------------
__global__ __launch_bounds__(256) void pad_kernel(const float* __restrict__ Wp2,
                                                  const float* __restrict__ bp2,
                                                  float* __restrict__ Wp,
                                                  float* __restrict__ bp) {
    int t = blockIdx.x * blockDim.x + threadIdx.x;
    if (t < DIM * OUT_PAD) {
        int k = t / OUT_PAD, n = t % OUT_PAD;
        Wp[t] = (n < OUT_DIM) ? Wp2[k * OUT_DIM + n] : 0.0f;
    } else if (t < DIM * OUT_PAD + OUT_PAD) {
        int j = t - DIM * OUT_PAD;
        bp[j] = (j < OUT_DIM) ? bp2[j] : 0.0f;
    }
}

// ---------------- row-wise log_softmax over 40 valid cols -----------------
__global__ __launch_bounds__(256) void logsoftmax_kernel(
    const float* __restrict__ o, float* __restrict__ out, int nrows) {
    int r = blockIdx.x * blockDim.x + threadIdx.x;
    if (r >= nrows) return;
    const float* p = o + (long)r * OUT_PAD;
    float v[OUT_DIM];
    float mx = -INFINITY;
#pragma unroll
    for (int c = 0; c < OUT_DIM; ++c) {
        v[c] = p[c];
        mx = fmaxf(mx, v[c]);
    }
    float s = 0.0f;
#pragma unroll
    for (int c = 0; c < OUT_DIM; ++c) s += expf(v[c] - mx);
    float l = logf(s);
    float* q = out + (long)r * OUT_DIM;
#pragma unroll
    for (int c = 0; c < OUT_DIM; ++c) q[c] = v[c] - mx - l;
}

extern "C" void kernel_launch(void* const* d_in, const int* in_sizes, int n_in,
                              void* d_out, int out_size, void* d_ws,
                              size_t ws_size, hipStream_t stream) {
    (void)in_sizes; (void)n_in; (void)out_size; (void)ws_size;

    const float* x   = (const float*)d_in[0];
    const int*   ei  = (const int*)d_in[1];  // [2,E]: src then dst
    const float* W1  = (const float*)d_in[2];
    const float* b1  = (const float*)d_in[3];
    const float* W2  = (const float*)d_in[4];
    const float* b2  = (const float*)d_in[5];
    const float* Wp1 = (const float*)d_in[6];
    const float* bp1 = (const float*)d_in[7];
    const float* Wp2 = (const float*)d_in[8];
    const float* bp2 = (const float*)d_in[9];

    const int* src = ei;
    const int* dst = ei + N_EDGES;

    // workspace layout
    float* bufA = (float*)d_ws;                          // N*64 (h buffers)
    float* bufB = bufA + (size_t)N_NODES * DIM;          // N*64 (agg / o)
    int*   degI = (int*)(bufB + (size_t)N_NODES * DIM);  // N
    float* Wp   = (float*)(degI + N_NODES);              // 64*48
    float* bp   = Wp + DIM * OUT_PAD;                    // 48

    const int MB = N_NODES / 16;  // 3125 M-tiles (N divisible by 16)
    const int SCAT_BLOCKS = (int)(((long)N_EDGES * 16 + 255) / 256);

    hipMemsetAsync(degI, 0, N_NODES * sizeof(int), stream);
    hipMemsetAsync(bufB, 0, (size_t)N_NODES * DIM * sizeof(float), stream);
    deg_kernel<<<(N_EDGES + 255) / 256, 256, 0, stream>>>(src, degI, N_EDGES);
    pad_kernel<<<(DIM * OUT_PAD + OUT_PAD + 255) / 256, 256, 0, stream>>>(
        Wp2, bp2, Wp, bp);

    // layer 1: h1 = x @ W1 + b1
    gemm_wmma<DIM><<<MB, 128, 0, stream>>>(x, W1, b1, nullptr, bufA);
    // agg1[src] += h1[dst]
    scatter_kernel<<<SCAT_BLOCKS, 256, 0, stream>>>(src, dst, bufA, bufB, N_EDGES);
    // layer 2: h2 = relu(agg1/deg) @ W2 + b2   (scale+relu fused into A load)
    gemm_wmma<DIM><<<MB, 128, 0, stream>>>(bufB, W2, b2, degI, bufA);
    // agg2[src] += h2[dst]
    hipMemsetAsync(bufB, 0, (size_t)N_NODES * DIM * sizeof(float), stream);
    scatter_kernel<<<SCAT_BLOCKS, 256, 0, stream>>>(src, dst, bufA, bufB, N_EDGES);
    // post_mp: h3 = relu(agg2/deg) @ Wp1 + bp1
    gemm_wmma<DIM><<<MB, 128, 0, stream>>>(bufB, Wp1, bp1, degI, bufA);
    // o = h3 @ Wp2pad + bp2pad  (N x 48, into bufB)
    gemm_wmma<OUT_PAD><<<MB, 96, 0, stream>>>(bufA, Wp, bp, nullptr, bufB);
    // log_softmax over 40 cols -> d_out
    logsoftmax_kernel<<<(N_NODES + 255) / 256, 256, 0, stream>>>(
        bufB, (float*)d_out, N_NODES);
}